// TransitionDown_70153995813101
// MI455X (gfx1250) — compile-verified
//
#include <hip/hip_runtime.h>

typedef __attribute__((ext_vector_type(2))) float v2f;
typedef __attribute__((ext_vector_type(8))) float v8f;

#define NPTS   8192
#define NPOINT 2048
#define KNN    16
#define CIN    64
#define COUT   128
#define BATCH  16

// ---------------- K0: zero the stats region of workspace ----------------
__global__ void k_zero(float* p, int n) {
  int i = blockIdx.x * blockDim.x + threadIdx.x;
  if (i < n) p[i] = 0.0f;
}

// ---------------- K1: farthest point sampling (1 block per batch) -------
__global__ __launch_bounds__(1024)
void k_fps(const float* __restrict__ xyz, float* __restrict__ new_xyz) {
  extern __shared__ float sm[];
  float* xs   = sm;                 // 8192
  float* ys   = sm + NPTS;          // 8192
  float* zs   = sm + 2 * NPTS;      // 8192
  float* dist = sm + 3 * NPTS;      // 8192   (128KB dynamic LDS total)
  __shared__ float rv[1024];
  __shared__ int   ri[1024];

  const int b   = blockIdx.x;
  const int tid = threadIdx.x;
  const float* base = xyz + (size_t)b * NPTS * 3;
  for (int i = tid; i < NPTS; i += 1024) {
    xs[i] = base[i * 3 + 0];
    ys[i] = base[i * 3 + 1];
    zs[i] = base[i * 3 + 2];
    dist[i] = 1e10f;
  }
  __syncthreads();

  int far = 0;
  for (int it = 0; it < NPOINT; ++it) {
    float cx = xs[far], cy = ys[far], cz = zs[far];   // LDS broadcast
    if (tid == 0) {
      float* o = new_xyz + ((size_t)b * NPOINT + it) * 3;
      o[0] = cx; o[1] = cy; o[2] = cz;
    }
    // update running min-distance, track local argmax
    float bestv = -1.0f; int besti = 0;
    for (int i = tid; i < NPTS; i += 1024) {
      float dx = xs[i] - cx, dy = ys[i] - cy, dz = zs[i] - cz;
      float d  = dx * dx + dy * dy + dz * dz;
      float nd = fminf(dist[i], d);
      dist[i] = nd;
      if (nd > bestv) { bestv = nd; besti = i; }
    }
    rv[tid] = bestv; ri[tid] = besti;
    __syncthreads();
    for (int s = 512; s > 0; s >>= 1) {
      if (tid < s) {
        float v2 = rv[tid + s]; int i2 = ri[tid + s];
        float v1 = rv[tid];     int i1 = ri[tid];
        if (v2 > v1 || (v2 == v1 && i2 < i1)) { rv[tid] = v2; ri[tid] = i2; }
      }
      __syncthreads();
    }
    far = ri[0];
    __syncthreads();
  }
}

// ---------------- K2: kNN (1 thread per query, xyz in LDS) --------------
__global__ __launch_bounds__(256)
void k_knn(const float* __restrict__ xyz, const float* __restrict__ new_xyz,
           int* __restrict__ knn_idx) {
  extern __shared__ float sm[];
  float* xs = sm; float* ys = sm + NPTS; float* zs = sm + 2 * NPTS; // 96KB
  const int b = blockIdx.y;
  const int m = blockIdx.x * 256 + threadIdx.x;
  const float* base = xyz + (size_t)b * NPTS * 3;
  for (int i = threadIdx.x; i < NPTS; i += 256) {
    xs[i] = base[i * 3 + 0];
    ys[i] = base[i * 3 + 1];
    zs[i] = base[i * 3 + 2];
  }
  __syncthreads();

  const float* q = new_xyz + ((size_t)b * NPOINT + m) * 3;
  float qx = q[0], qy = q[1], qz = q[2];

  float bd[KNN]; int bi[KNN];
#pragma unroll
  for (int j = 0; j < KNN; ++j) { bd[j] = 3.4e38f; bi[j] = 0; }

  for (int i = 0; i < NPTS; ++i) {
    float dx = xs[i] - qx, dy = ys[i] - qy, dz = zs[i] - qz;
    float d  = dx * dx + dy * dy + dz * dz;
    if (d < bd[KNN - 1]) {
      // unrolled sorted insert (stable: ties keep earlier index first)
#pragma unroll
      for (int j = KNN - 1; j >= 0; --j) {
        bool lt  = d < bd[j];
        bool ltp = (j > 0) ? (d < bd[j - 1]) : false;
        float nv = lt ? (ltp ? bd[j - 1] : d) : bd[j];
        int   ni = lt ? (ltp ? bi[j - 1] : i) : bi[j];
        bd[j] = nv; bi[j] = ni;
      }
    }
  }
  int* op = knn_idx + (size_t)(b * NPOINT + m) * KNN;
#pragma unroll
  for (int j = 0; j < KNN; ++j) op[j] = bi[j];
}

// ---- shared GEMM tile: 16 samples (one query's K neighbors) x 128 out ----
// A frag (16x4 f32): lane L holds sample (L&15), cols k0+2*(L>>4)+{0,1}
// B frag (4x16 f32): lane L holds col  nt*16+(L&15), rows mirror A's K split
__device__ __forceinline__ void gemm_tile(const float* __restrict__ feats,
                                          const int* __restrict__ knn_idx,
                                          const float* w_lds,
                                          int t, int lane, v8f acc[8]) {
  const int b    = t >> 11;                         // t = b*2048 + m
  const int* ip  = knn_idx + ((size_t)t << 4);
  const int row  = ip[lane & 15];
  const int half = lane >> 4;
  const float* grow  = feats + ((size_t)b * NPTS + row) * CIN + 2 * half;
  const float* wbase = w_lds + (lane & 15) * CIN + 2 * half;
#pragma unroll
  for (int ks = 0; ks < 16; ++ks) {
    v2f A = *(const v2f*)(grow + ks * 4);
#pragma unroll
    for (int nt = 0; nt < 8; ++nt) {
      v2f Bf = *(const v2f*)(wbase + nt * 16 * CIN + ks * 4);
      acc[nt] = __builtin_amdgcn_wmma_f32_16x16x4_f32(
          false, A, false, Bf, (short)0, acc[nt], false, false);
    }
  }
}

// ---------------- K3: GEMM pass A -> per-channel sum / sumsq ------------
__global__ __launch_bounds__(128)
void k_stats(const float* __restrict__ feats, const float* __restrict__ W,
             const float* __restrict__ bias, const int* __restrict__ knn_idx,
             float* __restrict__ sum, float* __restrict__ sumsq) {
  __shared__ float w_lds[COUT * CIN];               // 32KB, W[o][c] row-major
  for (int i = threadIdx.x; i < COUT * CIN; i += 128) w_lds[i] = W[i];
  __syncthreads();

  const int lane  = threadIdx.x & 31;
  const int wave  = threadIdx.x >> 5;
  const int tile0 = (blockIdx.x * 4 + wave) * 16;   // 16 tiles per wave

  float bo[8];
#pragma unroll
  for (int nt = 0; nt < 8; ++nt) bo[nt] = bias[nt * 16 + (lane & 15)];

  float csum[8] = {0.f, 0.f, 0.f, 0.f, 0.f, 0.f, 0.f, 0.f};
  float csq [8] = {0.f, 0.f, 0.f, 0.f, 0.f, 0.f, 0.f, 0.f};

  for (int tt = 0; tt < 16; ++tt) {
    v8f acc[8] = {};
    gemm_tile(feats, knn_idx, w_lds, tile0 + tt, lane, acc);
#pragma unroll
    for (int nt = 0; nt < 8; ++nt) {
      float s = 0.f, q = 0.f;
#pragma unroll
      for (int r = 0; r < 8; ++r) {
        float v = acc[nt][r] + bo[nt];
        s += v; q += v * v;
      }
      s += __shfl_xor(s, 16, 32);   // combine rows 0-7 with rows 8-15
      q += __shfl_xor(q, 16, 32);
      csum[nt] += s; csq[nt] += q;
    }
  }
  if (lane < 16) {
#pragma unroll
    for (int nt = 0; nt < 8; ++nt) {
      atomicAdd(&sum  [nt * 16 + lane], csum[nt]);
      atomicAdd(&sumsq[nt * 16 + lane], csq [nt]);
    }
  }
}

// ---------------- K4: fold BN stats into per-channel a, c ---------------
__global__ void k_bn(const float* __restrict__ sum, const float* __restrict__ sumsq,
                     const float* __restrict__ gamma, const float* __restrict__ beta,
                     float* __restrict__ avec, float* __restrict__ cvec) {
  int o = threadIdx.x;
  if (o < COUT) {
    const float inv = 1.0f / 524288.0f;             // B*M*K
    float mean = sum[o] * inv;
    float var  = fmaxf(sumsq[o] * inv - mean * mean, 0.0f);
    float a    = gamma[o] * rsqrtf(var + 1e-5f);
    avec[o] = a;
    cvec[o] = beta[o] - mean * a;
  }
}

// ---------------- K5: GEMM pass B -> BN + ReLU + max over K -------------
__global__ __launch_bounds__(128)
void k_final(const float* __restrict__ feats, const float* __restrict__ W,
             const float* __restrict__ bias, const float* __restrict__ avec,
             const float* __restrict__ cvec, const int* __restrict__ knn_idx,
             float* __restrict__ out_feat) {
  __shared__ float w_lds[COUT * CIN];
  for (int i = threadIdx.x; i < COUT * CIN; i += 128) w_lds[i] = W[i];
  __syncthreads();

  const int lane  = threadIdx.x & 31;
  const int wave  = threadIdx.x >> 5;
  const int tile0 = (blockIdx.x * 4 + wave) * 16;

  float ao[8], co[8];
#pragma unroll
  for (int nt = 0; nt < 8; ++nt) {
    int o = nt * 16 + (lane & 15);
    float a = avec[o];
    ao[nt] = a;
    co[nt] = cvec[o] + bias[o] * a;                 // fold linear bias into BN shift
  }

  for (int tt = 0; tt < 16; ++tt) {
    int t = tile0 + tt;
    v8f acc[8] = {};
    gemm_tile(feats, knn_idx, w_lds, t, lane, acc);
#pragma unroll
    for (int nt = 0; nt < 8; ++nt) {
      float mx = -3.4e38f;
#pragma unroll
      for (int r = 0; r < 8; ++r)
        mx = fmaxf(mx, fmaf(acc[nt][r], ao[nt], co[nt]));
      mx = fmaxf(mx, __shfl_xor(mx, 16, 32));       // rows 0-7 vs rows 8-15
      mx = fmaxf(mx, 0.0f);                         // ReLU (commutes with max)
      if (lane < 16)
        out_feat[(size_t)t * COUT + nt * 16 + lane] = mx;
    }
  }
}

extern "C" void kernel_launch(void* const* d_in, const int* in_sizes, int n_in,
                              void* d_out, int out_size, void* d_ws, size_t ws_size,
                              hipStream_t stream) {
  const float* xyz   = (const float*)d_in[0];
  const float* feats = (const float*)d_in[1];
  const float* W     = (const float*)d_in[2];
  const float* bias  = (const float*)d_in[3];
  const float* gamma = (const float*)d_in[4];
  const float* beta  = (const float*)d_in[5];

  float* out      = (float*)d_out;
  float* new_xyz  = out;                                   // [B, M, 3]
  float* new_feat = out + (size_t)BATCH * NPOINT * 3;      // [B, M, Cout]

  float* ws_f  = (float*)d_ws;
  float* sum   = ws_f;          // 128
  float* sumsq = ws_f + 128;    // 128
  float* avec  = ws_f + 256;    // 128
  float* cvec  = ws_f + 384;    // 128
  int*   knn_i = (int*)(ws_f + 512);                       // [B*M*K] = 2MB

  k_zero<<<1, 256, 0, stream>>>(ws_f, 256);
  k_fps<<<BATCH, 1024, 4 * NPTS * sizeof(float), stream>>>(xyz, new_xyz);
  k_knn<<<dim3(NPOINT / 256, BATCH), 256, 3 * NPTS * sizeof(float), stream>>>(
      xyz, new_xyz, knn_i);
  k_stats<<<512, 128, 0, stream>>>(feats, W, bias, knn_i, sum, sumsq);
  k_bn<<<1, 128, 0, stream>>>(sum, sumsq, gamma, beta, avec, cvec);
  k_final<<<512, 128, 0, stream>>>(feats, W, bias, avec, cvec, knn_i, new_feat);
}